// FSMNTeacher_29094108463603
// MI455X (gfx1250) — compile-verified
//
#include <hip/hip_runtime.h>

typedef __attribute__((ext_vector_type(2))) float v2f;
typedef __attribute__((ext_vector_type(8))) float v8f;

#define WMMA_F32X4(A, Bv, C) \
  __builtin_amdgcn_wmma_f32_16x16x4_f32(false, (A), false, (Bv), (short)0, (C), false, false)

constexpr int Bn = 4096, Tn = 128, SD = 512, CD = 128;
constexpr int CH = 48, KK = 5, HID = 64, OUTN = 2;
constexpr int ZD = CH + CD;     // 176
// padded LDS row strides (word-stride % 16 == 2 -> conflict-free, even -> b64 aligned)
constexpr int SDP  = SD + 2;    // 514
constexpr int CHP  = CH + 2;    // 50
constexpr int ZDP  = ZD + 2;    // 178
constexpr int HIDP = HID + 2;   // 66

// ---------------------------------------------------------------------------
// Kernel 1: h5[b*5+j, c] = relu(seq[b, 123+j, :] @ Wp + bp)[c]
// 20480 rows x 512 K x 48 N.  One wave => 16 rows x all 48 cols.
// Wp staged TRANSPOSED in LDS: sWpT[c][k], so a lane's B pair (K,K+1) is one
// contiguous ds_load_b64 landing directly in an even-aligned VGPR pair.
// ---------------------------------------------------------------------------
__global__ __launch_bounds__(256) void fsmn_proj_tail(
    const float* __restrict__ seq, const float* __restrict__ Wp,
    const float* __restrict__ bp, float* __restrict__ h5)
{
  __shared__ float sWpT[CH * SDP];           // 48*514*4 = 96.4 KB
  for (int idx = threadIdx.x; idx < SD * CH; idx += 256) {
    const int k = idx / CH, c = idx - k * CH;   // coalesced global read
    sWpT[c * SDP + k] = Wp[idx];                // scattered LDS write (one-time)
  }
  __syncthreads();

  const int lane = threadIdx.x & 31;
  const int wave = threadIdx.x >> 5;
  const int hi = lane >> 4;
  const int m = lane & 15;
  const int tile = blockIdx.x * 8 + wave;       // 0..1279

  const int row = tile * 16 + m;                // (b, j) pair index
  const int b = row / KK;
  const int j = row - b * KK;
  const float* arow  = seq + ((size_t)b * Tn + (Tn - KK) + j) * SD;
  const float* wcol0 = &sWpT[m * SDP];
  const float* wcol1 = &sWpT[(16 + m) * SDP];
  const float* wcol2 = &sWpT[(32 + m) * SDP];

  v8f a0 = {}, a1 = {}, a2 = {};
  #pragma unroll 4
  for (int k = 0; k < SD; k += 4) {
    const int kb = k + 2 * hi;
    const v2f av = *(const v2f*)(arow + kb);    // A[m][kb..kb+1]  (b64 global)
    const v2f b0 = *(const v2f*)(wcol0 + kb);   // B[kb..kb+1][m]  (b64 LDS)
    const v2f b1 = *(const v2f*)(wcol1 + kb);
    const v2f b2 = *(const v2f*)(wcol2 + kb);
    a0 = WMMA_F32X4(av, b0, a0);
    a1 = WMMA_F32X4(av, b1, a1);
    a2 = WMMA_F32X4(av, b2, a2);
  }

  const float bp0 = bp[m], bp1 = bp[16 + m], bp2 = bp[32 + m];
  #pragma unroll
  for (int i = 0; i < 8; ++i) {
    const size_t r = (size_t)tile * 16 + i + 8 * hi;
    float v0 = a0[i] + bp0; v0 = v0 > 0.f ? v0 : 0.f;
    float v1 = a1[i] + bp1; v1 = v1 > 0.f ? v1 : 0.f;
    float v2 = a2[i] + bp2; v2 = v2 > 0.f ? v2 : 0.f;
    h5[r * CH + m]      = v0;
    h5[r * CH + 16 + m] = v1;
    h5[r * CH + 32 + m] = v2;
  }
}

// ---------------------------------------------------------------------------
// Kernel 2: per wave, 16 batches:  mem -> h2 -> z -> hid -> out
// Wmix / W1 staged transposed+padded; all WMMA operand pairs are b64 loads.
// ---------------------------------------------------------------------------
__global__ __launch_bounds__(128) void fsmn_head(
    const float* __restrict__ h5, const float* __restrict__ ctx,
    const float* __restrict__ wmem, const float* __restrict__ bmem,
    const float* __restrict__ Wmix, const float* __restrict__ bmix,
    const float* __restrict__ W1, const float* __restrict__ b1,
    const float* __restrict__ W2, const float* __restrict__ b2,
    float* __restrict__ out)
{
  __shared__ float sWmixT[CH * CHP];      // 48*50*4   =  9.4 KB   [n][k]
  __shared__ float sW1T[HID * ZDP];       // 64*178*4  = 44.5 KB   [n][k]
  __shared__ float smem[4][16 * CHP];     // 12.5 KB
  __shared__ float sz[4][16 * ZDP];       // 44.5 KB
  __shared__ float shid[4][16 * HIDP];    // 16.5 KB

  for (int idx = threadIdx.x; idx < CH * CH; idx += 128) {
    const int k = idx / CH, n = idx - k * CH;
    sWmixT[n * CHP + k] = Wmix[idx];
  }
  for (int idx = threadIdx.x; idx < ZD * HID; idx += 128) {
    const int k = idx / HID, n = idx - k * HID;
    sW1T[n * ZDP + k] = W1[idx];
  }
  __syncthreads();

  const int lane = threadIdx.x & 31;
  const int wave = threadIdx.x >> 5;
  const int hi = lane >> 4;
  const int m = lane & 15;
  const int b0 = (blockIdx.x * 4 + wave) * 16;

  // ---- mem[mb,c] = bmem[c] + sum_j h5[(b0+mb)*5+j, c] * wmem[j,c] ----
  for (int idx = lane; idx < 16 * CH; idx += 32) {
    const int mb = idx / CH, c = idx - mb * CH;
    const size_t rbase = (size_t)(b0 + mb) * KK * CH + c;
    float acc = bmem[c];
    #pragma unroll
    for (int j = 0; j < KK; ++j)
      acc += h5[rbase + (size_t)j * CH] * wmem[j * CH + c];
    smem[wave][mb * CHP + c] = acc;
  }
  __syncthreads();

  // ---- h2 = relu(mem @ Wmix + bmix) -> z[:, 0:48] ----
  {
    v8f c0 = {}, c1 = {}, c2 = {};
    const float* wc0 = &sWmixT[m * CHP];
    const float* wc1 = &sWmixT[(16 + m) * CHP];
    const float* wc2 = &sWmixT[(32 + m) * CHP];
    const float* arow = &smem[wave][m * CHP];
    #pragma unroll
    for (int k = 0; k < CH; k += 4) {
      const int kb = k + 2 * hi;
      const v2f av = *(const v2f*)(arow + kb);
      const v2f w0 = *(const v2f*)(wc0 + kb);
      const v2f w1 = *(const v2f*)(wc1 + kb);
      const v2f w2 = *(const v2f*)(wc2 + kb);
      c0 = WMMA_F32X4(av, w0, c0);
      c1 = WMMA_F32X4(av, w1, c1);
      c2 = WMMA_F32X4(av, w2, c2);
    }
    const float q0 = bmix[m], q1 = bmix[16 + m], q2 = bmix[32 + m];
    #pragma unroll
    for (int i = 0; i < 8; ++i) {
      const int r = i + 8 * hi;
      float v0 = c0[i] + q0; v0 = v0 > 0.f ? v0 : 0.f;
      float v1 = c1[i] + q1; v1 = v1 > 0.f ? v1 : 0.f;
      float v2 = c2[i] + q2; v2 = v2 > 0.f ? v2 : 0.f;
      sz[wave][r * ZDP + m]      = v0;
      sz[wave][r * ZDP + 16 + m] = v1;
      sz[wave][r * ZDP + 32 + m] = v2;
    }
  }
  // ---- ctx -> z[:, 48:176] ----
  for (int idx = lane; idx < 16 * CD; idx += 32) {
    const int mb = idx >> 7, c = idx & (CD - 1);
    sz[wave][mb * ZDP + CH + c] = ctx[(size_t)(b0 + mb) * CD + c];
  }
  __syncthreads();

  // ---- hid = relu(z @ W1 + b1)  (M=16, K=176, N=64) ----
  {
    v8f h0 = {}, h1 = {}, h2 = {}, h3 = {};
    const float* wc0 = &sW1T[m * ZDP];
    const float* wc1 = &sW1T[(16 + m) * ZDP];
    const float* wc2 = &sW1T[(32 + m) * ZDP];
    const float* wc3 = &sW1T[(48 + m) * ZDP];
    const float* arow = &sz[wave][m * ZDP];
    #pragma unroll 4
    for (int k = 0; k < ZD; k += 4) {
      const int kb = k + 2 * hi;
      const v2f av = *(const v2f*)(arow + kb);
      const v2f w0 = *(const v2f*)(wc0 + kb);
      const v2f w1 = *(const v2f*)(wc1 + kb);
      const v2f w2 = *(const v2f*)(wc2 + kb);
      const v2f w3 = *(const v2f*)(wc3 + kb);
      h0 = WMMA_F32X4(av, w0, h0);
      h1 = WMMA_F32X4(av, w1, h1);
      h2 = WMMA_F32X4(av, w2, h2);
      h3 = WMMA_F32X4(av, w3, h3);
    }
    const float p0 = b1[m], p1 = b1[16 + m], p2 = b1[32 + m], p3 = b1[48 + m];
    #pragma unroll
    for (int i = 0; i < 8; ++i) {
      const int r = i + 8 * hi;
      float v0 = h0[i] + p0; v0 = v0 > 0.f ? v0 : 0.f;
      float v1 = h1[i] + p1; v1 = v1 > 0.f ? v1 : 0.f;
      float v2 = h2[i] + p2; v2 = v2 > 0.f ? v2 : 0.f;
      float v3 = h3[i] + p3; v3 = v3 > 0.f ? v3 : 0.f;
      shid[wave][r * HIDP + m]      = v0;
      shid[wave][r * HIDP + 16 + m] = v1;
      shid[wave][r * HIDP + 32 + m] = v2;
      shid[wave][r * HIDP + 48 + m] = v3;
    }
  }
  __syncthreads();

  // ---- out = hid @ W2 + b2  (one lane per (batch, out) pair; 16*2 = 32) ----
  {
    const int mb = lane >> 1, o = lane & 1;
    float acc = b2[o];
    #pragma unroll
    for (int q = 0; q < HID; ++q)
      acc += shid[wave][mb * HIDP + q] * W2[q * OUTN + o];
    out[(size_t)(b0 + mb) * OUTN + o] = acc;
  }
}

extern "C" void kernel_launch(void* const* d_in, const int* in_sizes, int n_in,
                              void* d_out, int out_size, void* d_ws, size_t ws_size,
                              hipStream_t stream) {
  const float* seq  = (const float*)d_in[0];
  const float* ctx  = (const float*)d_in[1];
  // d_in[2] = tab (unused by the reference's live outputs)
  const float* Wp   = (const float*)d_in[3];
  const float* bp   = (const float*)d_in[4];
  const float* wmem = (const float*)d_in[5];
  const float* bmem = (const float*)d_in[6];
  const float* Wmix = (const float*)d_in[7];
  const float* bmix = (const float*)d_in[8];
  const float* W1   = (const float*)d_in[9];
  const float* b1   = (const float*)d_in[10];
  const float* W2   = (const float*)d_in[11];
  const float* b2   = (const float*)d_in[12];

  float* h5  = (float*)d_ws;            // 20480 x 48 floats = 3.93 MB scratch
  float* out = (float*)d_out;

  // 1280 row-tiles of 16, 8 waves per block
  fsmn_proj_tail<<<dim3(160), dim3(256), 0, stream>>>(seq, Wp, bp, h5);
  // 256 batch-tiles of 16, 4 waves per block
  fsmn_head<<<dim3(64), dim3(128), 0, stream>>>(h5, ctx, wmem, bmem, Wmix, bmix,
                                                W1, b1, W2, b2, out);
}